// PointerNetwork_41867341201935
// MI455X (gfx1250) — compile-verified
//
#include <hip/hip_runtime.h>
#include <hip/hip_bf16.h>

// Problem constants (match reference)
#define BB 8
#define SRC 512
#define TGT 64
#define DD 512
#define NH 8
#define DH 64
#define VOC 32000
#define EXTV 2048
#define VT (VOC + EXTV)   // 34048

#define FMINV (-3.4028234663852886e38f)
#define EPSV  (1.1920928955078125e-07f)

typedef float v2f __attribute__((ext_vector_type(2)));
typedef float v8f __attribute__((ext_vector_type(8)));

// ---------------------------------------------------------------------------
// GEMM: C[M x 512] = A[M x 512] @ W[512 x 512] + bias[512]
// One wave per 16(M) x 64(N) strip: 4 fp32 WMMA accumulators, A fragment
// reused across the 4 N tiles. fp32 WMMA 16x16x4 fragment layout:
//   A: lane half 0 -> cols (k,k+1), half 1 -> cols (k+2,k+3), row = m0+(lane&15)
//   B: mirrored (rows k+2*half, k+2*half+1; col = n + (lane&15))
//   C VGPR i: row m0 + i + 8*half, col n + (lane&15)
// ---------------------------------------------------------------------------
__global__ __launch_bounds__(32) void gemm_wmma_f32(
    const float* __restrict__ A, const float* __restrict__ W,
    const float* __restrict__ bias, float* __restrict__ C)
{
    const int n0   = blockIdx.x * 64;
    const int m0   = blockIdx.y * 16;
    const int lane = threadIdx.x;
    const int half = lane >> 4;
    const int l16  = lane & 15;
    const int koff = half * 2;
    const float* arp = A + (size_t)(m0 + l16) * DD;

    v8f acc[4] = {};
    for (int k = 0; k < DD; k += 4) {
        v2f a = *(const v2f*)(arp + k + koff);
        const float* w0 = W + (size_t)(k + koff) * DD + n0 + l16;
#pragma unroll
        for (int j = 0; j < 4; ++j) {
            v2f b;
            b.x = w0[j * 16];
            b.y = w0[DD + j * 16];
            acc[j] = __builtin_amdgcn_wmma_f32_16x16x4_f32(
                false, a, false, b, (short)0, acc[j], false, false);
        }
    }
#pragma unroll
    for (int j = 0; j < 4; ++j) {
        const int bcol = n0 + j * 16 + l16;
        const float bv = bias[bcol];
#pragma unroll
        for (int i = 0; i < 8; ++i) {
            int row = m0 + i + half * 8;
            C[(size_t)row * DD + bcol] = acc[j][i] + bv;
        }
    }
}

// ---------------------------------------------------------------------------
// Scores: per (b,h): Sc[t][s] = (Q[b,t,h*64:] . K[b,s,h*64:]) / 8, pad -> -inf
// One wave per 16(T) x 64(S) strip; Q fragment reused across 4 S tiles.
// Both fragments are contiguous float2 loads (K-dim contiguous in memory).
// ---------------------------------------------------------------------------
__global__ __launch_bounds__(32) void scores_wmma(
    const float* __restrict__ Q, const float* __restrict__ K,
    const unsigned char* __restrict__ pad, float* __restrict__ Sc)
{
    const int s0   = blockIdx.x * 64;
    const int t0   = blockIdx.y * 16;
    const int bh   = blockIdx.z;           // b*8 + h
    const int b    = bh >> 3;
    const int h    = bh & 7;
    const int lane = threadIdx.x;
    const int half = lane >> 4;
    const int l16  = lane & 15;
    const int koff = half * 2;

    const float* qrow = Q + ((size_t)(b * TGT + t0 + l16)) * DD + h * DH;
    const float* krow = K + ((size_t)(b * SRC + s0 + l16)) * DD + h * DH;

    v8f acc[4] = {};
    for (int k = 0; k < DH; k += 4) {
        v2f a = *(const v2f*)(qrow + k + koff);
#pragma unroll
        for (int j = 0; j < 4; ++j) {
            v2f bf = *(const v2f*)(krow + (size_t)j * 16 * DD + k + koff);
            acc[j] = __builtin_amdgcn_wmma_f32_16x16x4_f32(
                false, a, false, bf, (short)0, acc[j], false, false);
        }
    }
    const float NEGINF = -__builtin_inff();
#pragma unroll
    for (int j = 0; j < 4; ++j) {
        const int scol = s0 + j * 16 + l16;
        const bool p = pad[b * SRC + scol] != 0;
#pragma unroll
        for (int i = 0; i < 8; ++i) {
            int t = t0 + i + half * 8;
            float v = p ? NEGINF : acc[j][i] * 0.125f;   // 1/sqrt(64)
            Sc[((size_t)bh * TGT + t) * SRC + scol] = v;
        }
    }
}

// ---------------------------------------------------------------------------
// Softmax over S per (b,h,t), then mean over heads -> Attn (B,T,S)
// ---------------------------------------------------------------------------
__global__ __launch_bounds__(512) void softmax_mean(
    const float* __restrict__ Sc, float* __restrict__ Attn)
{
    __shared__ float red[512];
    const int bt = blockIdx.x;            // b*64 + t
    const int b  = bt >> 6;
    const int t  = bt & 63;
    const int s  = threadIdx.x;
    const float NEGINF = -__builtin_inff();

    float acc = 0.f;
    for (int h = 0; h < NH; ++h) {
        float v = Sc[(((size_t)(b * NH + h)) * TGT + t) * SRC + s];
        red[s] = v;
        __syncthreads();
        for (int off = 256; off > 0; off >>= 1) {
            if (s < off) red[s] = fmaxf(red[s], red[s + off]);
            __syncthreads();
        }
        float mx = red[0];
        __syncthreads();
        float e = 0.f;
        if (mx > NEGINF) e = __expf(v - mx);   // v==-inf -> 0
        red[s] = e;
        __syncthreads();
        for (int off = 256; off > 0; off >>= 1) {
            if (s < off) red[s] += red[s + off];
            __syncthreads();
        }
        float sum = red[0];
        __syncthreads();
        if (sum > 0.f) acc += e / (sum * (float)NH);
    }
    Attn[(size_t)bt * SRC + s] = acc;
}

// ---------------------------------------------------------------------------
// Owner slot: owner[b,s] = first s' with ids[b,s'] == ids[b,s]
// (dedups repeated vocab ids so the sparse fix-up is race-free)
// ---------------------------------------------------------------------------
__global__ void compute_owner(const int* __restrict__ ids, int* __restrict__ owner)
{
    int idx = blockIdx.x * blockDim.x + threadIdx.x;   // over B*SRC
    if (idx >= BB * SRC) return;
    int b = idx >> 9;
    int s = idx & (SRC - 1);
    int my = ids[idx];
    int o = s;
    for (int j = 0; j < s; ++j) {
        if (ids[b * SRC + j] == my) { o = j; break; }
    }
    owner[idx] = o;
}

// ---------------------------------------------------------------------------
// Aggregate attention per owner slot: agg[b,t,owner] += attn[b,t,s]
// ---------------------------------------------------------------------------
__global__ void aggregate_attn(const float* __restrict__ attn,
                               const int* __restrict__ owner,
                               float* __restrict__ agg)
{
    int idx = blockIdx.x * blockDim.x + threadIdx.x;   // over B*T*S
    if (idx >= BB * TGT * SRC) return;
    int s  = idx & (SRC - 1);
    int bt = idx >> 9;
    int b  = bt >> 6;
    int o  = owner[b * SRC + s];
    atomicAdd(agg + (size_t)bt * SRC + o, attn[idx]);
}

// ---------------------------------------------------------------------------
// Zero fill (agg buffer, 1 MB)
// ---------------------------------------------------------------------------
__global__ void fill_zero(float* __restrict__ p, long n) {
    long i = (long)blockIdx.x * blockDim.x + threadIdx.x;
    if (i < n) p[i] = 0.f;
}

// ---------------------------------------------------------------------------
// Copy gate: z = tgt[b,t,:] . w_copy + b_copy -> log(cp), log(1-cp)
// ---------------------------------------------------------------------------
__device__ __forceinline__ float softplusf(float x) {
    if (x > 20.f)  return x;
    if (x < -20.f) return __expf(x);
    return log1pf(__expf(x));
}

__global__ __launch_bounds__(64) void copy_gate(
    const float* __restrict__ tgt, const float* __restrict__ w,
    const float* __restrict__ bptr,
    float* __restrict__ log_cp, float* __restrict__ log_1mcp)
{
    __shared__ float red[64];
    const int bt = blockIdx.x;
    float sum = 0.f;
    for (int d = threadIdx.x; d < DD; d += 64)
        sum += tgt[(size_t)bt * DD + d] * w[d];
    red[threadIdx.x] = sum;
    __syncthreads();
    for (int off = 32; off > 0; off >>= 1) {
        if (threadIdx.x < off) red[threadIdx.x] += red[threadIdx.x + off];
        __syncthreads();
    }
    if (threadIdx.x == 0) {
        float z  = red[0] + bptr[0];
        float sp = softplusf(-z);
        log_cp[bt]   = -sp;          // log(sigmoid(z))
        log_1mcp[bt] = -z - sp;      // log(1 - sigmoid(z))
    }
}

// ---------------------------------------------------------------------------
// Per-(b,t) logsumexp over the 32000 logits
// ---------------------------------------------------------------------------
__global__ __launch_bounds__(256) void row_lse(
    const float* __restrict__ logits, float* __restrict__ lse)
{
    __shared__ float red[256];
    const int bt = blockIdx.x;
    const float* row = logits + (size_t)bt * VOC;
    const int tid = threadIdx.x;

    float mx = -__builtin_inff();
    for (int v = tid; v < VOC; v += 256) mx = fmaxf(mx, row[v]);
    red[tid] = mx;
    __syncthreads();
    for (int off = 128; off > 0; off >>= 1) {
        if (tid < off) red[tid] = fmaxf(red[tid], red[tid + off]);
        __syncthreads();
    }
    mx = red[0];
    __syncthreads();

    float sum = 0.f;
    for (int v = tid; v < VOC; v += 256) sum += __expf(row[v] - mx);
    red[tid] = sum;
    __syncthreads();
    for (int off = 128; off > 0; off >>= 1) {
        if (tid < off) red[tid] += red[tid + off];
        __syncthreads();
    }
    if (tid == 0) lse[bt] = mx + __logf(red[0]);
}

// ---------------------------------------------------------------------------
// Dense pass: assume att == 0 everywhere -> p1 = log(eps) + log(cp) per row.
// grid = (ceil(VT/256), B*T)
// ---------------------------------------------------------------------------
__global__ __launch_bounds__(256) void final_dense(
    const float* __restrict__ logits, const float* __restrict__ lse,
    const float* __restrict__ lcp, const float* __restrict__ l1mcp,
    float* __restrict__ out)
{
    const int v  = blockIdx.x * 256 + threadIdx.x;
    const int bt = blockIdx.y;
    if (v >= VT) return;

    float p1 = __logf(EPSV) + lcp[bt];
    float p0 = (v < VOC) ? (logits[(size_t)bt * VOC + v] - lse[bt] + l1mcp[bt])
                         : FMINV;
    float m = fmaxf(p0, p1);
    float n = fminf(p0, p1);
    out[(size_t)bt * VT + v] = m + log1pf(__expf(n - m));
}

// ---------------------------------------------------------------------------
// Sparse fix-up: owner slots only; overwrite out[b,t,ids[b,s]] with the
// correct logaddexp using aggregated attention mass. Each (b,t,v) is
// touched by exactly one owner slot -> race-free.
// ---------------------------------------------------------------------------
__global__ void fixup_scatter(
    const int* __restrict__ ids, const int* __restrict__ owner,
    const float* __restrict__ agg, const float* __restrict__ logits,
    const float* __restrict__ lse, const float* __restrict__ lcp,
    const float* __restrict__ l1mcp, float* __restrict__ out)
{
    int idx = blockIdx.x * blockDim.x + threadIdx.x;   // over B*T*S
    if (idx >= BB * TGT * SRC) return;
    int s  = idx & (SRC - 1);
    int bt = idx >> 9;
    int b  = bt >> 6;
    if (owner[b * SRC + s] != s) return;

    float a  = agg[(size_t)bt * SRC + s];
    int   v  = ids[b * SRC + s];
    float p1 = __logf(a == 0.f ? EPSV : a) + lcp[bt];
    float p0 = (v < VOC) ? (logits[(size_t)bt * VOC + v] - lse[bt] + l1mcp[bt])
                         : FMINV;
    float m = fmaxf(p0, p1);
    float n = fminf(p0, p1);
    out[(size_t)bt * VT + v] = m + log1pf(__expf(n - m));
}

// ---------------------------------------------------------------------------
extern "C" void kernel_launch(void* const* d_in, const int* in_sizes, int n_in,
                              void* d_out, int out_size, void* d_ws, size_t ws_size,
                              hipStream_t stream) {
    const float*         logits  = (const float*)d_in[0];
    const int*           ids     = (const int*)d_in[1];
    const float*         src     = (const float*)d_in[2];
    const unsigned char* padmask = (const unsigned char*)d_in[3]; // jax bool = 1 byte
    const float*         tgt     = (const float*)d_in[4];
    const float*         w_q     = (const float*)d_in[5];
    const float*         b_q     = (const float*)d_in[6];
    const float*         w_k     = (const float*)d_in[7];
    const float*         b_k     = (const float*)d_in[8];
    const float*         w_copy  = (const float*)d_in[9];
    const float*         b_copy  = (const float*)d_in[10];
    float*               out     = (float*)d_out;

    // Workspace layout (256-byte aligned slabs)
    char* ws = (char*)d_ws;
    size_t off = 0;
    auto take = [&](size_t bytes) -> char* {
        char* p = ws + off;
        off += (bytes + 255) & ~(size_t)255;
        return p;
    };
    float* qbuf  = (float*)take((size_t)BB * TGT * DD * 4);        // 1 MB
    float* kbuf  = (float*)take((size_t)BB * SRC * DD * 4);        // 8 MB
    float* sc    = (float*)take((size_t)BB * NH * TGT * SRC * 4);  // 8 MB
    float* attn  = (float*)take((size_t)BB * TGT * SRC * 4);       // 1 MB
    float* agg   = (float*)take((size_t)BB * TGT * SRC * 4);       // 1 MB
    int*   owner = (int*)  take((size_t)BB * SRC * 4);
    float* lse   = (float*)take((size_t)BB * TGT * 4);
    float* lcp   = (float*)take((size_t)BB * TGT * 4);
    float* l1mcp = (float*)take((size_t)BB * TGT * 4);

    // Zero the (small) aggregation buffer
    const long agg_n = (long)BB * TGT * SRC;
    fill_zero<<<(int)((agg_n + 255) / 256), 256, 0, stream>>>(agg, agg_n);

    // Q projection: M = B*T = 512; K projection: M = B*S = 4096 (WMMA fp32)
    gemm_wmma_f32<<<dim3(DD / 64, (BB * TGT) / 16), 32, 0, stream>>>(tgt, w_q, b_q, qbuf);
    gemm_wmma_f32<<<dim3(DD / 64, (BB * SRC) / 16), 32, 0, stream>>>(src, w_k, b_k, kbuf);

    // Attention scores (masked, scaled) via WMMA
    scores_wmma<<<dim3(SRC / 64, TGT / 16, BB * NH), 32, 0, stream>>>(qbuf, kbuf, padmask, sc);

    // Softmax over S, mean over heads
    softmax_mean<<<BB * TGT, SRC, 0, stream>>>(sc, attn);

    // Dedup ids, aggregate attention per owner slot
    compute_owner<<<(BB * SRC + 255) / 256, 256, 0, stream>>>(ids, owner);
    aggregate_attn<<<(BB * TGT * SRC + 255) / 256, 256, 0, stream>>>(attn, owner, agg);

    // Copy gate + logits LSE
    copy_gate<<<BB * TGT, 64, 0, stream>>>(tgt, w_copy, b_copy, lcp, l1mcp);
    row_lse<<<BB * TGT, 256, 0, stream>>>(logits, lse);

    // Dense output pass (att treated as 0), then sparse fix-up of touched slots
    final_dense<<<dim3((VT + 255) / 256, BB * TGT), 256, 0, stream>>>(
        logits, lse, lcp, l1mcp, out);
    fixup_scatter<<<(BB * TGT * SRC + 255) / 256, 256, 0, stream>>>(
        ids, owner, agg, logits, lse, lcp, l1mcp, out);
}